// SAGE_5454608466092
// MI455X (gfx1250) — compile-verified
//
#include <hip/hip_runtime.h>

#define NNODES 100000
#define NEDGES 3200000
#define D 64

typedef __attribute__((ext_vector_type(2))) float v2f;
typedef __attribute__((ext_vector_type(8))) float v8f;

__global__ void zero_f32(float* __restrict__ p, long long n) {
  long long i = (long long)blockIdx.x * blockDim.x + threadIdx.x;
  long long stride = (long long)gridDim.x * blockDim.x;
  for (; i < n; i += stride) p[i] = 0.0f;
}

// degree = segment_sum(ones, dst) ; layer-invariant, computed once
__global__ void count_deg(const long long* __restrict__ dst,
                          float* __restrict__ cnt, int E) {
  int e = blockIdx.x * blockDim.x + threadIdx.x;
  if (e < E) atomicAdd(&cnt[(int)dst[e]], 1.0f);
}

// One wave per edge, 2 floats per lane: agg[dst] += x[src]
__global__ __launch_bounds__(256) void scatter_feats(
    const float* __restrict__ xin, const long long* __restrict__ src,
    const long long* __restrict__ dst, float* __restrict__ agg, int E) {
  int e = blockIdx.x * 8 + (threadIdx.x >> 5);   // uniform within wave
  int lane = threadIdx.x & 31;
  if (e >= E) return;
  int s = (int)src[e];
  int d = (int)dst[e];
  const float2 v = *(const float2*)(xin + (size_t)s * D + lane * 2);
  float* p = agg + (size_t)d * D + lane * 2;
  atomicAdd(p, v.x);
  atomicAdd(p + 1, v.y);
}

// out = (agg * 1/max(deg,1)) @ Wl^T + bl + xin @ Wr^T
// Block: 256 threads = 8 waves, 32 nodes. Wave (w): node-tile = w>>2, j-tile = w&3.
// WMMA f32 16x16x4, K=64 in 16 steps, two A/B pairs chained into one accumulator.
__global__ __launch_bounds__(256) void sage_combine(
    const float* __restrict__ xin, const float* __restrict__ agg,
    const float* __restrict__ cnt, const float* __restrict__ Wl,
    const float* __restrict__ bl, const float* __restrict__ Wr,
    float* __restrict__ out, int N) {
  __shared__ float sA[32 * D];   // mean-aggregated neighbor features
  __shared__ float sX[32 * D];   // self features
  const int nodeBlock = blockIdx.x * 32;

  for (int i = threadIdx.x; i < 32 * D; i += 256) {
    int r = i >> 6, c = i & 63;
    int node = nodeBlock + r;
    if (node < N) {
      float inv = 1.0f / fmaxf(cnt[node], 1.0f);
      sA[i] = agg[(size_t)node * D + c] * inv;
      sX[i] = xin[(size_t)node * D + c];
    } else {
      sA[i] = 0.0f;
      sX[i] = 0.0f;
    }
  }
  __syncthreads();

  const int lane  = threadIdx.x & 31;
  const int wave  = threadIdx.x >> 5;
  const int tileR = (wave >> 2) * 16;   // row tile within block: 0 or 16
  const int jbase = (wave & 3) * 16;    // output-column tile
  const int half  = lane >> 4;          // 0 -> K{0,1}, 1 -> K{2,3}
  const int l15   = lane & 15;
  const int rA    = tileR + l15;        // A-matrix row this lane supplies
  const int colW  = jbase + l15;        // B-matrix column this lane supplies

  v8f acc = {0.f, 0.f, 0.f, 0.f, 0.f, 0.f, 0.f, 0.f};
#pragma unroll
  for (int kk = 0; kk < 16; ++kk) {
    const int k0 = kk * 4 + half * 2;
    v2f a1, b1, a2, b2;
    a1.x = sA[rA * D + k0];      a1.y = sA[rA * D + k0 + 1];
    b1.x = Wl[colW * D + k0];    b1.y = Wl[colW * D + k0 + 1];  // B[k][j] = Wl[j][k]
    a2.x = sX[rA * D + k0];      a2.y = sX[rA * D + k0 + 1];
    b2.x = Wr[colW * D + k0];    b2.y = Wr[colW * D + k0 + 1];
    acc = __builtin_amdgcn_wmma_f32_16x16x4_f32(false, a1, false, b1,
                                                (short)0, acc, false, false);
    acc = __builtin_amdgcn_wmma_f32_16x16x4_f32(false, a2, false, b2,
                                                (short)0, acc, false, false);
  }

  const float bias = bl[jbase + l15];
#pragma unroll
  for (int v = 0; v < 8; ++v) {
    int m = half ? (v + 8) : v;            // C/D layout: lanes>=16 hold M=v+8
    int node = nodeBlock + tileR + m;
    if (node < N) out[(size_t)node * D + jbase + l15] = acc[v] + bias;
  }
}

__global__ void init_out(float* __restrict__ out, const float* __restrict__ blin) {
  out[0] = blin[0];
}

// d_out += (1/N) * sum_n dot(h[n,:], Wlin)  -- wave per node, shfl reduction
__global__ __launch_bounds__(256) void pool_dot(const float* __restrict__ h,
                                                const float* __restrict__ Wlin,
                                                float* __restrict__ out, int N) {
  int gwave  = (blockIdx.x * blockDim.x + threadIdx.x) >> 5;
  int nwaves = (gridDim.x * blockDim.x) >> 5;
  int lane   = threadIdx.x & 31;
  float2 wv = *(const float2*)(Wlin + lane * 2);
  float partial = 0.0f;
  for (int node = gwave; node < N; node += nwaves) {
    float2 hv = *(const float2*)(h + (size_t)node * D + lane * 2);
    partial += hv.x * wv.x + hv.y * wv.y;
  }
  for (int off = 16; off > 0; off >>= 1)
    partial += __shfl_xor(partial, off, 32);
  if (lane == 0) atomicAdd(out, partial * (1.0f / (float)N));
}

extern "C" void kernel_launch(void* const* d_in, const int* in_sizes, int n_in,
                              void* d_out, int out_size, void* d_ws, size_t ws_size,
                              hipStream_t stream) {
  (void)in_sizes; (void)n_in; (void)out_size; (void)ws_size;
  const float*     x    = (const float*)d_in[0];
  const long long* ei   = (const long long*)d_in[1];   // int64 [2, E]
  const long long* src  = ei;
  const long long* dst  = ei + NEDGES;
  const float* Wl1 = (const float*)d_in[2];
  const float* bl1 = (const float*)d_in[3];
  const float* Wr1 = (const float*)d_in[4];
  const float* Wl2 = (const float*)d_in[5];
  const float* bl2 = (const float*)d_in[6];
  const float* Wr2 = (const float*)d_in[7];
  const float* Wl3 = (const float*)d_in[8];
  const float* bl3 = (const float*)d_in[9];
  const float* Wr3 = (const float*)d_in[10];
  const float* Wlin = (const float*)d_in[11];
  const float* blin = (const float*)d_in[12];
  float* outp = (float*)d_out;

  float* ws = (float*)d_ws;
  const size_t AGG_OFF = 100096;                         // cnt padded to 64-float mult
  const size_t HA_OFF  = AGG_OFF + (size_t)NNODES * D;
  const size_t HB_OFF  = HA_OFF + (size_t)NNODES * D;
  float* cnt = ws;
  float* agg = ws + AGG_OFF;
  float* hA  = ws + HA_OFF;
  float* hB  = ws + HB_OFF;

  dim3 b256(256);
  const int combineBlocks = (NNODES + 31) / 32;          // 3125
  const int scatterBlocks = (NEDGES + 7) / 8;            // 400000

  // degrees (shared by all layers)
  zero_f32<<<dim3(512), b256, 0, stream>>>(cnt, (long long)NNODES);
  count_deg<<<dim3((NEDGES + 255) / 256), b256, 0, stream>>>(dst, cnt, NEDGES);

  // layer 1: x -> hA
  zero_f32<<<dim3(2048), b256, 0, stream>>>(agg, (long long)NNODES * D);
  scatter_feats<<<dim3(scatterBlocks), b256, 0, stream>>>(x, src, dst, agg, NEDGES);
  sage_combine<<<dim3(combineBlocks), b256, 0, stream>>>(x, agg, cnt, Wl1, bl1, Wr1, hA, NNODES);

  // layer 2: hA -> hB
  zero_f32<<<dim3(2048), b256, 0, stream>>>(agg, (long long)NNODES * D);
  scatter_feats<<<dim3(scatterBlocks), b256, 0, stream>>>(hA, src, dst, agg, NEDGES);
  sage_combine<<<dim3(combineBlocks), b256, 0, stream>>>(hA, agg, cnt, Wl2, bl2, Wr2, hB, NNODES);

  // layer 3: hB -> hA
  zero_f32<<<dim3(2048), b256, 0, stream>>>(agg, (long long)NNODES * D);
  scatter_feats<<<dim3(scatterBlocks), b256, 0, stream>>>(hB, src, dst, agg, NEDGES);
  sage_combine<<<dim3(combineBlocks), b256, 0, stream>>>(hB, agg, cnt, Wl3, bl3, Wr3, hA, NNODES);

  // global mean pool + final linear -> scalar
  init_out<<<dim3(1), dim3(1), 0, stream>>>(outp, blin);
  pool_dot<<<dim3(512), b256, 0, stream>>>(hA, Wlin, outp, NNODES);
}